// QuantumMambaLayer_74431783239901
// MI455X (gfx1250) — compile-verified
//
#include <hip/hip_runtime.h>
#include <math.h>

// ---------------------------------------------------------------------------
// QuantumMambaLayer for MI455X (gfx1250, wave32)
//   K1: x_proj = x @ W_in + b_in        -> f32 WMMA 16x16x4, 16x64 slab/wave
//   K2: LayerNorm + L2 normalize rows   -> in-place in workspace
//   K3: 3 quantum circuits (state in LDS) + measure + combine + tiny MLP
// ---------------------------------------------------------------------------

#define N_QUBITS   10
#define STATE_DIM  1024
#define FEATURE    1024
#define BATCH      1024
#define MEAS_DIM   30
#define OUT_DIM    64

typedef float v2f __attribute__((ext_vector_type(2)));
typedef float v8f __attribute__((ext_vector_type(8)));

struct cplx { float x, y; };
__device__ __forceinline__ cplx cmk(float a, float b) { cplx c; c.x = a; c.y = b; return c; }
__device__ __forceinline__ cplx cmul(cplx a, cplx b) { return cmk(a.x*b.x - a.y*b.y, a.x*b.y + a.y*b.x); }
__device__ __forceinline__ cplx cadd(cplx a, cplx b) { return cmk(a.x + b.x, a.y + b.y); }

// ---------------------------------------------------------------------------
// Kernel 1: 1024x1024x1024 f32 GEMM with V_WMMA_F32_16X16X4_F32.
// One wave -> 16x64 slab of C (4 accumulators share the A fragment).
// 4 waves (128 threads) per block; 64 M-tiles x 16 N-groups = 1024 waves.
// A layout (16x4):  lanes 0-15 M=lane hold K={0,1}; lanes 16-31 M=lane-16 hold K={2,3}
// B layout (4x16):  lanes hold N=lane&15; reg0=K{0|2}, reg1=K{1|3} by lane half
// C/D layout:       reg j -> M = j + (lane>=16 ? 8 : 0), N = lane&15
// ---------------------------------------------------------------------------
__global__ void __launch_bounds__(128)
gemm_wmma_f32(const float* __restrict__ x, const float* __restrict__ W,
              const float* __restrict__ bias, float* __restrict__ out)
{
    const int lane = threadIdx.x & 31;
    const int wave = threadIdx.x >> 5;
    const int wt   = blockIdx.x * 4 + wave;    // 1024 wave-slabs
    const int tm   = wt >> 4;                  // 0..63  (M tile)
    const int tg   = wt & 15;                  // 0..15  (N group of 64)
    const int m    = lane & 15;
    const int half = lane >> 4;
    const int kb   = half << 1;                // 0 or 2 (K sub-offset)
    const int row  = tm * 16 + m;
    const int col0 = tg * 64 + m;              // tiles at col0 + {0,16,32,48}

    const float* __restrict__ arow = x + (size_t)row * FEATURE;

    v8f acc0 = {}, acc1 = {}, acc2 = {}, acc3 = {};

#pragma unroll 2
    for (int k = 0; k < FEATURE; k += 4) {
        // Branchless prefetch of the B stream 16 K-steps ahead
        // (wrapped with &1023 to stay in-bounds -> no per-iteration branch).
        const int kpf = (k + 16) & (FEATURE - 1);
        __builtin_prefetch(&W[(size_t)(kpf + kb) * STATE_DIM + col0], 0, 1);

        v2f a;
        a[0] = arow[k + kb];
        a[1] = arow[k + kb + 1];

        const float* __restrict__ wr0 = W + (size_t)(k + kb) * STATE_DIM;
        const float* __restrict__ wr1 = wr0 + STATE_DIM;

        v2f b0, b1, b2, b3;
        b0[0] = wr0[col0];      b0[1] = wr1[col0];
        b1[0] = wr0[col0 + 16]; b1[1] = wr1[col0 + 16];
        b2[0] = wr0[col0 + 32]; b2[1] = wr1[col0 + 32];
        b3[0] = wr0[col0 + 48]; b3[1] = wr1[col0 + 48];

        acc0 = __builtin_amdgcn_wmma_f32_16x16x4_f32(false, a, false, b0, (short)0, acc0, false, false);
        acc1 = __builtin_amdgcn_wmma_f32_16x16x4_f32(false, a, false, b1, (short)0, acc1, false, false);
        acc2 = __builtin_amdgcn_wmma_f32_16x16x4_f32(false, a, false, b2, (short)0, acc2, false, false);
        acc3 = __builtin_amdgcn_wmma_f32_16x16x4_f32(false, a, false, b3, (short)0, acc3, false, false);
    }

    const float bv0 = bias[col0];
    const float bv1 = bias[col0 + 16];
    const float bv2 = bias[col0 + 32];
    const float bv3 = bias[col0 + 48];
    const int mbase = tm * 16 + (half << 3);
#pragma unroll
    for (int j = 0; j < 8; ++j) {
        float* o = out + (size_t)(mbase + j) * STATE_DIM + col0;
        o[0]  = acc0[j] + bv0;
        o[16] = acc1[j] + bv1;
        o[32] = acc2[j] + bv2;
        o[48] = acc3[j] + bv3;
    }
}

// ---------------------------------------------------------------------------
// Block reduction helper (256 threads = 8 waves, wave32 shuffles)
// ---------------------------------------------------------------------------
__device__ __forceinline__ float block_reduce_256(float v, float* scratch, int tid)
{
#pragma unroll
    for (int off = 16; off > 0; off >>= 1) v += __shfl_xor(v, off, 32);
    if ((tid & 31) == 0) scratch[tid >> 5] = v;
    __syncthreads();
    float r = scratch[0] + scratch[1] + scratch[2] + scratch[3] +
              scratch[4] + scratch[5] + scratch[6] + scratch[7];
    __syncthreads();
    return r;
}

// ---------------------------------------------------------------------------
// Kernel 2: per-row LayerNorm then L2 normalize (in place). 1 block = 1 row.
// ---------------------------------------------------------------------------
__global__ void __launch_bounds__(256)
ln_l2_kernel(float* __restrict__ buf, const float* __restrict__ g,
             const float* __restrict__ bb)
{
    __shared__ float scratch[8];
    const int tid = threadIdx.x;
    float* row = buf + (size_t)blockIdx.x * STATE_DIM;

    float v[4], s = 0.f;
#pragma unroll
    for (int j = 0; j < 4; ++j) { v[j] = row[tid + j * 256]; s += v[j]; }
    const float mu = block_reduce_256(s, scratch, tid) * (1.f / STATE_DIM);

    float sq = 0.f;
#pragma unroll
    for (int j = 0; j < 4; ++j) { float d = v[j] - mu; sq += d * d; }
    const float var  = block_reduce_256(sq, scratch, tid) * (1.f / STATE_DIM);
    const float rstd = rsqrtf(var + 1e-5f);

    float xn[4], n2 = 0.f;
#pragma unroll
    for (int j = 0; j < 4; ++j) {
        int c = tid + j * 256;
        xn[j] = (v[j] - mu) * rstd * g[c] + bb[c];
        n2 += xn[j] * xn[j];
    }
    const float N2  = block_reduce_256(n2, scratch, tid);
    const float inv = 1.f / fmaxf(sqrtf(N2), 1e-12f);
#pragma unroll
    for (int j = 0; j < 4; ++j) row[tid + j * 256] = xn[j] * inv;
}

// ---------------------------------------------------------------------------
// Kernel 3: quantum circuits in LDS (8KB state), one block per batch item.
// 512 threads = 512 amplitude pairs per gate. Qubit w pairs indices differing
// by stride 2^(9-w) (reference reshape convention: w=0 is the MSB).
// ---------------------------------------------------------------------------
__device__ __forceinline__ void load_state(cplx* st, const float* __restrict__ base,
                                           int b, int tid)
{
    __syncthreads();   // previous readers of st must be done
    st[tid]       = cmk(base[(size_t)b * STATE_DIM + tid], 0.f);
    st[tid + 512] = cmk(base[(size_t)b * STATE_DIM + tid + 512], 0.f);
}

__device__ __forceinline__ void gate1(cplx* st, int w,
                                      cplx m00, cplx m01, cplx m10, cplx m11, int tid)
{
    __syncthreads();
    const int stride = 1 << (9 - w);
    const int i0 = ((tid & ~(stride - 1)) << 1) | (tid & (stride - 1));
    const int i1 = i0 + stride;
    const cplx s0 = st[i0], s1 = st[i1];
    st[i0] = cadd(cmul(m00, s0), cmul(m01, s1));
    st[i1] = cadd(cmul(m10, s0), cmul(m11, s1));
}

__device__ __forceinline__ void cgate(cplx* st, int c, int t,
                                      cplx m00, cplx m01, cplx m10, cplx m11, int tid)
{
    __syncthreads();
    const int stride = 1 << (9 - t);
    const int i0 = ((tid & ~(stride - 1)) << 1) | (tid & (stride - 1));
    const int i1 = i0 + stride;
    const int cm = 1 << (9 - c);
    if (i0 & cm) {
        const cplx s0 = st[i0], s1 = st[i1];
        st[i0] = cadd(cmul(m00, s0), cmul(m01, s1));
        st[i1] = cadd(cmul(m10, s0), cmul(m11, s1));
    }
}

__device__ __forceinline__ void g_ry(cplx* st, int w, float t, int tid) {
    float c = cosf(0.5f * t), s = sinf(0.5f * t);
    gate1(st, w, cmk(c,0), cmk(-s,0), cmk(s,0), cmk(c,0), tid);
}
__device__ __forceinline__ void g_rz(cplx* st, int w, float t, int tid) {
    float c = cosf(0.5f * t), s = sinf(0.5f * t);
    gate1(st, w, cmk(c,-s), cmk(0,0), cmk(0,0), cmk(c,s), tid);
}
__device__ __forceinline__ void g_rx(cplx* st, int w, float t, int tid) {
    float c = cosf(0.5f * t), s = sinf(0.5f * t);
    gate1(st, w, cmk(c,0), cmk(0,-s), cmk(0,-s), cmk(c,0), tid);
}

__device__ __forceinline__ void measure_xyz(const cplx* st, float* meas /*30*/,
                                            float (*red)[3], int tid)
{
#pragma unroll 1
    for (int w = 0; w < N_QUBITS; ++w) {
        __syncthreads();
        const int stride = 1 << (9 - w);
        const int i0 = ((tid & ~(stride - 1)) << 1) | (tid & (stride - 1));
        const int i1 = i0 + stride;
        const cplx s0 = st[i0], s1 = st[i1];
        float cr = s0.x * s1.x + s0.y * s1.y;             // Re(conj(s0)*s1)
        float ci = s0.x * s1.y - s0.y * s1.x;             // Im(conj(s0)*s1)
        float ez = (s0.x*s0.x + s0.y*s0.y) - (s1.x*s1.x + s1.y*s1.y);
#pragma unroll
        for (int off = 16; off > 0; off >>= 1) {
            cr += __shfl_xor(cr, off, 32);
            ci += __shfl_xor(ci, off, 32);
            ez += __shfl_xor(ez, off, 32);
        }
        if ((tid & 31) == 0) {
            red[tid >> 5][0] = cr; red[tid >> 5][1] = ci; red[tid >> 5][2] = ez;
        }
        __syncthreads();
        if (tid == 0) {
            float a = 0.f, b = 0.f, c = 0.f;
#pragma unroll
            for (int i = 0; i < 16; ++i) { a += red[i][0]; b += red[i][1]; c += red[i][2]; }
            meas[w * 3 + 0] = 2.f * a;
            meas[w * 3 + 1] = 2.f * b;
            meas[w * 3 + 2] = c;
        }
    }
}

__global__ void __launch_bounds__(512)
quantum_kernel(const float* __restrict__ base,
               const float* __restrict__ qlcu, const float* __restrict__ bp,
               const float* __restrict__ cp,   const float* __restrict__ gp,
               const float* __restrict__ sp,
               const float* __restrict__ ar, const float* __restrict__ ai,
               const float* __restrict__ br, const float* __restrict__ bi,
               const float* __restrict__ gr, const float* __restrict__ gi,
               const float* __restrict__ W1, const float* __restrict__ b1,
               const float* __restrict__ W2, const float* __restrict__ b2,
               float* __restrict__ out)
{
    __shared__ cplx  st[STATE_DIM];
    __shared__ float red[16][3];
    __shared__ float meas[3][MEAS_DIM];
    __shared__ float comb[MEAS_DIM];
    __shared__ float hbuf[MEAS_DIM];

    const int tid = threadIdx.x;
    const int b   = blockIdx.x;

    // ---------- circuit 1: ssm_circuit ----------
    load_state(st, base, b, tid);
    int idx = 0;
#pragma unroll 1
    for (int layer = 0; layer < 2; ++layer) {
        for (int i = 0; i < N_QUBITS; ++i) {
            g_ry(st, i, qlcu[idx++], tid);
            g_rz(st, i, qlcu[idx++], tid);
        }
        for (int i = 0; i < N_QUBITS - 1; ++i) {
            float t = qlcu[idx++];
            float c = cosf(0.5f * t), s = sinf(0.5f * t);
            cgate(st, i, i + 1, cmk(c,0), cmk(0,-s), cmk(0,-s), cmk(c,0), tid);
        }
        for (int i = 0; i < N_QUBITS; ++i) g_ry(st, i, bp[i], tid);
        for (int i = 0; i < N_QUBITS; ++i) g_rx(st, i, qlcu[idx++], tid);
        for (int i = 0; i < N_QUBITS; ++i) g_rz(st, i, cp[i], tid);
    }
    measure_xyz(st, meas[0], red, tid);

    // ---------- circuit 2: gate_circuit ----------
    load_state(st, base, b, tid);
    for (int i = 0; i < N_QUBITS; ++i) g_ry(st, i, gp[i], tid);
    for (int i = 0; i < N_QUBITS - 1; ++i)
        cgate(st, i, i + 1, cmk(0,0), cmk(1,0), cmk(1,0), cmk(0,0), tid);   // CNOT
    measure_xyz(st, meas[1], red, tid);

    // ---------- circuit 3: skip_circuit ----------
    load_state(st, base, b, tid);
    for (int i = 0; i < N_QUBITS; ++i) g_rz(st, i, sp[i], tid);
    measure_xyz(st, meas[2], red, tid);

    // ---------- combine + normalize ----------
    __syncthreads();
    if (tid == 0) {
        const float arr = ar[0], aii = ai[0];
        const float brr = br[0], bii = bi[0];
        const float grr = gr[0], gii = gi[0];
        float n2 = 0.f;
        for (int j = 0; j < MEAS_DIM; ++j) {
            float pr = arr * meas[0][j] + brr * meas[1][j] + grr * meas[2][j];
            float pi = aii * meas[0][j] + bii * meas[1][j] + gii * meas[2][j];
            float m2 = pr * pr + pi * pi;
            comb[j] = sqrtf(m2);
            n2 += m2;
        }
        const float inv = 1.f / sqrtf(n2 + 1e-8f);
        for (int j = 0; j < MEAS_DIM; ++j) comb[j] *= inv;
    }
    __syncthreads();

    // ---------- tiny MLP tail ----------
    if (tid < MEAS_DIM) {
        float acc = b1[tid];
        for (int j = 0; j < MEAS_DIM; ++j) acc += comb[j] * W1[j * MEAS_DIM + tid];
        hbuf[tid] = fmaxf(acc, 0.f);
    }
    __syncthreads();
    if (tid < OUT_DIM) {
        float acc = b2[tid];
        for (int i = 0; i < MEAS_DIM; ++i) acc += hbuf[i] * W2[i * OUT_DIM + tid];
        out[(size_t)b * OUT_DIM + tid] = acc;
    }
}

// ---------------------------------------------------------------------------
extern "C" void kernel_launch(void* const* d_in, const int* in_sizes, int n_in,
                              void* d_out, int out_size, void* d_ws, size_t ws_size,
                              hipStream_t stream)
{
    const float* x      = (const float*)d_in[0];
    const float* W_in   = (const float*)d_in[1];
    const float* b_in   = (const float*)d_in[2];
    const float* ln_g   = (const float*)d_in[3];
    const float* ln_b   = (const float*)d_in[4];
    const float* qlcu   = (const float*)d_in[5];
    const float* bp     = (const float*)d_in[6];
    const float* cp     = (const float*)d_in[7];
    const float* gp     = (const float*)d_in[8];
    const float* sp     = (const float*)d_in[9];
    const float* ar     = (const float*)d_in[10];
    const float* ai     = (const float*)d_in[11];
    const float* br     = (const float*)d_in[12];
    const float* bi     = (const float*)d_in[13];
    const float* gr     = (const float*)d_in[14];
    const float* gi     = (const float*)d_in[15];
    const float* W1     = (const float*)d_in[16];
    const float* b1     = (const float*)d_in[17];
    const float* W2     = (const float*)d_in[18];
    const float* b2     = (const float*)d_in[19];

    float* xproj = (float*)d_ws;           // 1024*1024 f32 = 4 MB scratch

    // K1: GEMM (64 M-tiles x 16 N-groups, 4 wave-slabs per block)
    gemm_wmma_f32<<<256, 128, 0, stream>>>(x, W_in, b_in, xproj);
    // K2: LayerNorm + L2 normalize, in place
    ln_l2_kernel<<<BATCH, 256, 0, stream>>>(xproj, ln_g, ln_b);
    // K3: quantum circuits + measurement + MLP, fused
    quantum_kernel<<<BATCH, 512, 0, stream>>>(xproj, qlcu, bp, cp, gp, sp,
                                              ar, ai, br, bi, gr, gi,
                                              W1, b1, W2, b2, (float*)d_out);
}